// Encoder_70282844831870
// MI455X (gfx1250) — compile-verified
//
#include <hip/hip_runtime.h>
#include <math.h>

#define N_NODES 50000
#define N_EDGES 600000
#define DIM     128
#define BN_EPS  1e-5f

#define TM          64          // rows per block (4 waves x 16)
#define LDS_STRIDE  132         // 128 + 4 pad -> conflict-free b64 A-fragment reads

typedef __attribute__((ext_vector_type(2))) float v2f;
typedef __attribute__((ext_vector_type(8))) float v8f;

// ---------------------------------------------------------------------------
// Zero-fill (graph-capture-safe scratch init)
// ---------------------------------------------------------------------------
__global__ void zero_kernel(float* __restrict__ p, size_t n) {
    size_t i = (size_t)blockIdx.x * blockDim.x + threadIdx.x;
    if (i < n) p[i] = 0.0f;
}

// ---------------------------------------------------------------------------
// Repack one 128x128 weight matrix into WMMA B-fragment order:
//   Wf[(kb*8 + n)*32 + lane] = { W[k][c], W[k+1][c] },
//   k = 4*kb + 2*(lane>>4),  c = 16*n + (lane&15)
// -> inner-loop B load becomes ONE coalesced global_load_b64 per fragment.
// ---------------------------------------------------------------------------
__global__ void repack_w_kernel(float* __restrict__ dst, const float* __restrict__ W) {
    int idx = blockIdx.x * blockDim.x + threadIdx.x;   // 0 .. 8191
    if (idx >= 32 * 8 * 32) return;
    int lane = idx & 31;
    int n    = (idx >> 5) & 7;
    int kb   = idx >> 8;
    int k    = kb * 4 + ((lane >> 4) << 1);
    int c    = n * 16 + (lane & 15);
    float2 v;
    v.x = W[(size_t)k * DIM + c];
    v.y = W[(size_t)(k + 1) * DIM + c];
    *(float2*)(dst + (size_t)idx * 2) = v;
}

// ---------------------------------------------------------------------------
// Scatter-add: one wave32 per edge; each lane moves a float4 (32*4 = 128 floats)
// agg[dst[e]] += x[src[e]].  agg (25.6MB) is L2-resident -> L2 atomics.
// ---------------------------------------------------------------------------
__global__ void scatter_add_kernel(float* __restrict__ agg,
                                   const float* __restrict__ x,
                                   const int* __restrict__ src,
                                   const int* __restrict__ dst) {
    int wave = (int)(((size_t)blockIdx.x * blockDim.x + threadIdx.x) >> 5);
    int lane = threadIdx.x & 31;
    if (wave >= N_EDGES) return;
    int s = src[wave];
    int d = dst[wave];
    const float4 v = ((const float4*)(x + (size_t)s * DIM))[lane];
    float* a = agg + (size_t)d * DIM + lane * 4;
    atomicAdd(a + 0, v.x);
    atomicAdd(a + 1, v.y);
    atomicAdd(a + 2, v.z);
    atomicAdd(a + 3, v.w);
}

// ---------------------------------------------------------------------------
// GEMM1: t = relu((x + agg) @ W + b).
// Block = 128 threads (4 waves) = 64 output rows. x+agg fused during
// synchronous LDS staging (needs the add in registers); padded stride 132
// -> conflict-free ds_load_b64 A fragments. B: coalesced b64 from Wf.
// ---------------------------------------------------------------------------
__global__ void gemm1_kernel(float* __restrict__ t,
                             const float* __restrict__ x,
                             const float* __restrict__ agg,
                             const float* __restrict__ Wf,
                             const float* __restrict__ bias) {
    __shared__ float As[TM * LDS_STRIDE];
    const int tid  = threadIdx.x;
    const int row0 = blockIdx.x * TM;

    // Stage x+agg tile into LDS: 64 rows x 32 float4s, fully coalesced.
    for (int idx = tid; idx < TM * 32; idx += 128) {
        int r  = idx >> 5;
        int c4 = idx & 31;
        int gr = row0 + r;
        if (gr < N_NODES) {
            const float4 xv = *(const float4*)(x   + (size_t)gr * DIM + c4 * 4);
            const float4 gv = *(const float4*)(agg + (size_t)gr * DIM + c4 * 4);
            float4 s;
            s.x = xv.x + gv.x; s.y = xv.y + gv.y;
            s.z = xv.z + gv.z; s.w = xv.w + gv.w;
            *(float4*)(&As[r * LDS_STRIDE + c4 * 4]) = s;
        }
    }
    __syncthreads();

    const int wave  = tid >> 5;
    const int lane  = tid & 31;
    const int rw    = wave * 16;                 // wave's row offset in tile
    if (row0 + rw >= N_NODES) return;            // wave-uniform (50000 % 16 == 0)

    const int khalf = (lane >> 4) << 1;
    const int bcol  = lane & 15;
    const float* aplds = &As[(rw + (lane & 15)) * LDS_STRIDE + khalf];
    const v2f*   wfp   = (const v2f*)Wf + lane;

    v8f acc[8];
    #pragma unroll
    for (int n = 0; n < 8; ++n) acc[n] = (v8f)0.0f;

    for (int kb = 0; kb < 32; ++kb) {
        v2f a = *(const v2f*)(aplds + kb * 4);
        #pragma unroll
        for (int n = 0; n < 8; ++n) {
            v2f b = wfp[(kb * 8 + n) * 32];
            acc[n] = __builtin_amdgcn_wmma_f32_16x16x4_f32(
                false, a, false, b, (short)0, acc[n], false, false);
        }
    }

    const int rbase = row0 + rw + 8 * (lane >> 4);
    #pragma unroll
    for (int n = 0; n < 8; ++n) {
        int c = n * 16 + bcol;
        float bv = bias[c];
        #pragma unroll
        for (int i = 0; i < 8; ++i) {
            float v = acc[n][i] + bv;
            v = v > 0.0f ? v : 0.0f;             // MLP internal ReLU
            t[(size_t)(rbase + i) * DIM + c] = v;
        }
    }
}

// ---------------------------------------------------------------------------
// GEMM2: h = act(t @ W + b), act = relu (layers 0-2) or tanh (layers 3-6).
// Tile staging uses CDNA5 async memory->LDS DMA (global_load_async_to_lds_b128,
// ASYNCcnt-tracked, per-lane LDS address keeps the padded layout), then
// s_wait_asynccnt + barrier. Plus BatchNorm stats via LDS block reduction.
// ---------------------------------------------------------------------------
__global__ void gemm2_kernel(float* __restrict__ h,
                             const float* __restrict__ t,
                             const float* __restrict__ Wf,
                             const float* __restrict__ bias,
                             float* __restrict__ sums,
                             float* __restrict__ sumsq,
                             int use_tanh) {
    __shared__ float As[TM * LDS_STRIDE];
    __shared__ float lsum[DIM];
    __shared__ float lsq[DIM];
    const int tid  = threadIdx.x;
    const int row0 = blockIdx.x * TM;

    if (tid < DIM) { lsum[tid] = 0.0f; lsq[tid] = 0.0f; }

    // Async-stage the 64x128 input tile directly into LDS (no VGPR round trip).
    for (int idx = tid; idx < TM * 32; idx += 128) {
        int r  = idx >> 5;
        int c4 = idx & 31;
        int gr = row0 + r;
        if (gr < N_NODES) {     // per-lane EXEC mask gates the async issue
            unsigned lds_addr = (unsigned)(uintptr_t)(&As[r * LDS_STRIDE + c4 * 4]);
            unsigned long long gaddr =
                (unsigned long long)(uintptr_t)(t + (size_t)gr * DIM + c4 * 4);
            asm volatile("global_load_async_to_lds_b128 %0, %1, off"
                         :: "v"(lds_addr), "v"(gaddr) : "memory");
        }
    }
    asm volatile("s_wait_asynccnt 0x0" ::: "memory");
    __syncthreads();

    const int wave   = tid >> 5;
    const int lane   = tid & 31;
    const int rw     = wave * 16;
    const bool active = (row0 + rw < N_NODES);   // wave-uniform

    if (active) {
        const int khalf = (lane >> 4) << 1;
        const int bcol  = lane & 15;
        const float* aplds = &As[(rw + (lane & 15)) * LDS_STRIDE + khalf];
        const v2f*   wfp   = (const v2f*)Wf + lane;

        v8f acc[8];
        #pragma unroll
        for (int n = 0; n < 8; ++n) acc[n] = (v8f)0.0f;

        for (int kb = 0; kb < 32; ++kb) {
            v2f a = *(const v2f*)(aplds + kb * 4);
            #pragma unroll
            for (int n = 0; n < 8; ++n) {
                v2f b = wfp[(kb * 8 + n) * 32];
                acc[n] = __builtin_amdgcn_wmma_f32_16x16x4_f32(
                    false, a, false, b, (short)0, acc[n], false, false);
            }
        }

        const int rbase = row0 + rw + 8 * (lane >> 4);
        #pragma unroll
        for (int n = 0; n < 8; ++n) {
            int c = n * 16 + bcol;
            float bv = bias[c];
            float psum = 0.0f, psq = 0.0f;
            #pragma unroll
            for (int i = 0; i < 8; ++i) {
                float v = acc[n][i] + bv;
                v = use_tanh ? tanhf(v) : (v > 0.0f ? v : 0.0f);
                h[(size_t)(rbase + i) * DIM + c] = v;
                psum += v;
                psq  += v * v;
            }
            atomicAdd(&lsum[c], psum);
            atomicAdd(&lsq[c],  psq);
        }
    }
    __syncthreads();
    if (tid < DIM) {
        atomicAdd(&sums[tid],  lsum[tid]);
        atomicAdd(&sumsq[tid], lsq[tid]);
    }
}

// ---------------------------------------------------------------------------
// BN finalize: per-column scale/shift from accumulated stats (1 block).
// ---------------------------------------------------------------------------
__global__ void bn_finalize_kernel(float* __restrict__ scale,
                                   float* __restrict__ shift,
                                   const float* __restrict__ sums,
                                   const float* __restrict__ sumsq,
                                   const float* __restrict__ gamma,
                                   const float* __restrict__ beta) {
    int c = threadIdx.x;
    if (c < DIM) {
        const float inv_n = 1.0f / (float)N_NODES;
        float mean = sums[c] * inv_n;
        float var  = sumsq[c] * inv_n - mean * mean;
        float rstd = rsqrtf(var + BN_EPS);
        float sc   = gamma[c] * rstd;
        scale[c] = sc;
        shift[c] = beta[c] - mean * sc;
    }
}

// ---------------------------------------------------------------------------
// BN apply: y = h * scale[col] + shift[col], vectorized float4.
// ---------------------------------------------------------------------------
__global__ void bn_apply_kernel(float* __restrict__ out,
                                const float* __restrict__ h,
                                const float* __restrict__ scale,
                                const float* __restrict__ shift) {
    size_t i = (size_t)blockIdx.x * blockDim.x + threadIdx.x;
    size_t n4 = (size_t)N_NODES * DIM / 4;
    if (i >= n4) return;
    float4 v = ((const float4*)h)[i];
    int c = (int)((i * 4) & (DIM - 1));
    float4 o;
    o.x = v.x * scale[c + 0] + shift[c + 0];
    o.y = v.y * scale[c + 1] + shift[c + 1];
    o.z = v.z * scale[c + 2] + shift[c + 2];
    o.w = v.w * scale[c + 3] + shift[c + 3];
    ((float4*)out)[i] = o;
}

// ---------------------------------------------------------------------------
// Driver
// ---------------------------------------------------------------------------
extern "C" void kernel_launch(void* const* d_in, const int* in_sizes, int n_in,
                              void* d_out, int out_size, void* d_ws, size_t ws_size,
                              hipStream_t stream) {
    (void)in_sizes; (void)n_in; (void)out_size; (void)ws_size;

    const float* x_in  = (const float*)d_in[0];
    const int*   edge  = (const int*)d_in[1];
    const int*   src   = edge;
    const int*   dst   = edge + N_EDGES;
    // d_in[2] = batch (all zeros; unused)
    const float* W1 = (const float*)d_in[3];
    const float* b1 = (const float*)d_in[4];
    const float* W2 = (const float*)d_in[5];
    const float* b2 = (const float*)d_in[6];
    const float* gamma = (const float*)d_in[7];
    const float* beta  = (const float*)d_in[8];
    float* out = (float*)d_out;

    const size_t ND    = (size_t)N_NODES * DIM;
    const size_t WFSZ  = 32 * 8 * 32 * 2;        // floats per repacked matrix
    float* agg   = (float*)d_ws;
    float* tbuf  = agg  + ND;
    float* hbuf  = tbuf + ND;
    float* xcur  = hbuf + ND;
    float* stats = xcur + ND;
    float* sums  = stats;
    float* sumsq = stats + DIM;
    float* scale = stats + 2 * DIM;
    float* shift = stats + 3 * DIM;
    float* wf    = stats + 4 * DIM;              // 14 repacked matrices
    auto Wf1 = [&](int l) { return wf + (size_t)l * WFSZ; };
    auto Wf2 = [&](int l) { return wf + (size_t)(7 + l) * WFSZ; };

    const int SC_BLOCKS    = N_EDGES / 8;                    // 8 edges per block
    const int GEMM_BLOCKS  = (N_NODES + TM - 1) / TM;        // 782
    const int APPLY_BLOCKS = (int)((ND / 4 + 255) / 256);
    const int ZERO_BLOCKS  = (int)((ND + 255) / 256);
    const int RP_BLOCKS    = (32 * 8 * 32 + 255) / 256;      // 32

    // One-time weight repack (cheap; deterministic every call).
    for (int l = 0; l < 7; ++l) {
        repack_w_kernel<<<RP_BLOCKS, 256, 0, stream>>>(Wf1(l), W1 + (size_t)l * DIM * DIM);
        repack_w_kernel<<<RP_BLOCKS, 256, 0, stream>>>(Wf2(l), W2 + (size_t)l * DIM * DIM);
    }

    auto run_layer = [&](const float* xin, float* yout, int l, int use_tanh) {
        zero_kernel<<<1, 256, 0, stream>>>(stats, 2 * DIM);
        gemm1_kernel<<<GEMM_BLOCKS, 128, 0, stream>>>(
            tbuf, xin, agg, Wf1(l), b1 + (size_t)l * DIM);
        gemm2_kernel<<<GEMM_BLOCKS, 128, 0, stream>>>(
            hbuf, tbuf, Wf2(l), b2 + (size_t)l * DIM, sums, sumsq, use_tanh);
        bn_finalize_kernel<<<1, 128, 0, stream>>>(
            scale, shift, sums, sumsq,
            gamma + (size_t)l * DIM, beta + (size_t)l * DIM);
        bn_apply_kernel<<<APPLY_BLOCKS, 256, 0, stream>>>(yout, hbuf, scale, shift);
    };

    const float* x = x_in;
    // Layers 0..2: scatter + GIN + BN(relu)
    for (int l = 0; l < 3; ++l) {
        zero_kernel<<<ZERO_BLOCKS, 256, 0, stream>>>(agg, ND);
        scatter_add_kernel<<<SC_BLOCKS, 256, 0, stream>>>(agg, x, src, dst);
        run_layer(x, xcur, l, /*use_tanh=*/0);
        x = xcur;
    }
    // Layers 3..6 share the same input x -> compute the scatter-add ONCE.
    zero_kernel<<<ZERO_BLOCKS, 256, 0, stream>>>(agg, ND);
    scatter_add_kernel<<<SC_BLOCKS, 256, 0, stream>>>(agg, x, src, dst);
    for (int k = 0; k < 4; ++k) {
        run_layer(x, out + (size_t)k * ND, 3 + k, /*use_tanh=*/1);
    }
}